// OurLinker_15899968930392
// MI455X (gfx1250) — compile-verified
//
#include <hip/hip_runtime.h>

typedef __attribute__((ext_vector_type(2))) float v2f;
typedef __attribute__((ext_vector_type(4))) float v4f;
typedef __attribute__((ext_vector_type(8))) float v8f;

#define TAGS 2000
#define VIDS 20000
#define DIM  768

// ---------------------------------------------------------------------------
// 1) Non-temporal vectorized zero fill of the 320MB output (bandwidth floor).
//    NT stores keep the 192MB L2 free for h_video (61MB) gather reuse.
// ---------------------------------------------------------------------------
__global__ void zero_fill(float* __restrict__ out, long long n4) {
    long long i = (long long)blockIdx.x * blockDim.x + threadIdx.x;
    if (i < n4) {
        v4f z = {0.f, 0.f, 0.f, 0.f};
        __builtin_nontemporal_store(z, (v4f*)out + i);
    }
}

// ---------------------------------------------------------------------------
// 2) Edge scores via V_WMMA_F32_16X16X4_F32: each wave handles 16 edges.
//    A row m  = h_tag[src[g*16+m]]   (16x4 slice per step)
//    B col n  = h_video[dst[g*16+n]] (4x16 slice per step)
//    Edge score = diagonal D[n][n] of the accumulated 16x16 product.
//    EXEC is all-ones through the WMMA loop (wave-uniform early-exit only).
// ---------------------------------------------------------------------------
__global__ void edge_scores(const float* __restrict__ h_tag,
                            const float* __restrict__ h_video,
                            const int* __restrict__ pos_src,
                            const int* __restrict__ pos_dst,
                            const int* __restrict__ neg_src,
                            const int* __restrict__ neg_dst,
                            float* __restrict__ cls,
                            int pos_groups, int total_groups)
{
    unsigned tid  = blockIdx.x * blockDim.x + threadIdx.x;
    int      wave = (int)(tid >> 5);
    int      lane = (int)(threadIdx.x & 31u);
    if (wave >= total_groups) return;   // wave-uniform: EXEC stays full below

    const int* __restrict__ srcp;
    const int* __restrict__ dstp;
    int base;
    if (wave < pos_groups) { srcp = pos_src; dstp = pos_dst; base = wave * 16; }
    else                   { srcp = neg_src; dstp = neg_dst; base = (wave - pos_groups) * 16; }

    int m    = lane & 15;   // edge-in-group this lane serves (row of A == col of B)
    int half = lane >> 4;   // K-half: lanes 0-15 hold K={0,1}, lanes 16-31 K={2,3}

    int si = srcp[base + m];
    int di = dstp[base + m];

    const float* __restrict__ arow = h_tag   + (size_t)si * DIM + 2 * half;
    const float* __restrict__ brow = h_video + (size_t)di * DIM + 2 * half;

    v8f c = {};
    #pragma unroll 8
    for (int k0 = 0; k0 < DIM; k0 += 4) {
        v2f a = *(const v2f*)(arow + k0);   // A[m][k0+2h], A[m][k0+2h+1]
        v2f b = *(const v2f*)(brow + k0);   // B[2h][n],    B[2h+1][n]
        c = __builtin_amdgcn_wmma_f32_16x16x4_f32(
                /*neg_a=*/false, a, /*neg_b=*/false, b,
                /*c_mod=*/(short)0, c, /*reuse_a=*/false, /*reuse_b=*/false);
    }

    // Diagonal D[n][n]: rows 0-7 live in lanes 0-7 (vgpr = n),
    // rows 8-15 live in lanes 24-31 (vgpr = n-8). Those lanes already hold
    // edge n's (si, di) since n == lane&15 for them.
    bool holder = (lane < 8) || (lane >= 24);
    int  dn     = (lane < 8) ? lane : (lane - 24);
    float s = c[0];
    #pragma unroll
    for (int r = 1; r < 8; ++r) { if (dn == r) s = c[r]; }

    if (holder) {
        __hip_atomic_fetch_add(cls + (size_t)di * TAGS + si, s,
                               __ATOMIC_RELAXED, __HIP_MEMORY_SCOPE_AGENT);
    }
}

// Scalar tail (only launched if E % 16 != 0; with E=100000 it never runs).
__global__ void edge_tail(const float* __restrict__ h_tag,
                          const float* __restrict__ h_video,
                          const int* __restrict__ src,
                          const int* __restrict__ dst,
                          float* __restrict__ cls, int start, int count)
{
    int i = blockIdx.x * blockDim.x + threadIdx.x;
    if (i >= count) return;
    int si = src[start + i], di = dst[start + i];
    const float* a = h_tag   + (size_t)si * DIM;
    const float* b = h_video + (size_t)di * DIM;
    float s = 0.f;
    for (int k = 0; k < DIM; ++k) s += a[k] * b[k];
    __hip_atomic_fetch_add(cls + (size_t)di * TAGS + si, s,
                           __ATOMIC_RELAXED, __HIP_MEMORY_SCOPE_AGENT);
}

// ---------------------------------------------------------------------------
// 3) labels[pos_dst, pos_src] += 1.0
// ---------------------------------------------------------------------------
__global__ void labels_fill(const int* __restrict__ pos_src,
                            const int* __restrict__ pos_dst,
                            float* __restrict__ labels, int e)
{
    int i = blockIdx.x * blockDim.x + threadIdx.x;
    if (i >= e) return;
    __hip_atomic_fetch_add(labels + (size_t)pos_dst[i] * TAGS + pos_src[i], 1.0f,
                           __ATOMIC_RELAXED, __HIP_MEMORY_SCOPE_AGENT);
}

extern "C" void kernel_launch(void* const* d_in, const int* in_sizes, int n_in,
                              void* d_out, int out_size, void* d_ws, size_t ws_size,
                              hipStream_t stream)
{
    const float* h_tag   = (const float*)d_in[0];
    const float* h_video = (const float*)d_in[1];
    const int*   pos_src = (const int*)d_in[2];
    const int*   pos_dst = (const int*)d_in[3];
    const int*   neg_src = (const int*)d_in[4];
    const int*   neg_dst = (const int*)d_in[5];

    float* cls    = (float*)d_out;                       // [VIDS, TAGS]
    float* labels = cls + (size_t)VIDS * TAGS;           // [VIDS, TAGS]

    int e_pos = in_sizes[2];
    int e_neg = in_sizes[4];

    // 1) zero the whole output (out_size floats, multiple of 4)
    long long n4 = (long long)out_size / 4;
    unsigned zblocks = (unsigned)((n4 + 255) / 256);
    zero_fill<<<zblocks, 256, 0, stream>>>((float*)d_out, n4);

    // 2) WMMA edge scores (16 edges per wave, pos + neg fused in one grid)
    int pg = e_pos / 16;
    int ng = e_neg / 16;
    int total = pg + ng;
    if (total > 0) {
        int waves_per_block = 256 / 32;
        int blocks = (total + waves_per_block - 1) / waves_per_block;
        edge_scores<<<blocks, 256, 0, stream>>>(h_tag, h_video,
                                                pos_src, pos_dst,
                                                neg_src, neg_dst,
                                                cls, pg, total);
    }
    int pt = e_pos - pg * 16;
    if (pt > 0)
        edge_tail<<<1, 256, 0, stream>>>(h_tag, h_video, pos_src, pos_dst,
                                         cls, pg * 16, pt);
    int nt = e_neg - ng * 16;
    if (nt > 0)
        edge_tail<<<1, 256, 0, stream>>>(h_tag, h_video, neg_src, neg_dst,
                                         cls, ng * 16, nt);

    // 3) labels
    if (e_pos > 0)
        labels_fill<<<(e_pos + 255) / 256, 256, 0, stream>>>(pos_src, pos_dst,
                                                             labels, e_pos);
}